// GraphConvolution_40905268527669
// MI455X (gfx1250) — compile-verified
//
#include <hip/hip_runtime.h>

// GCN graph-conv fused pipeline for gfx1250 (wave32, WMMA, async-LDS).
//
//   PA  = X @ (W_A0 + W_A1)       (folds 2 GEMMs + 2 spmm passes into 1 each)
//   PB0 = X @ W_B0,  PB1 = X @ W_B1
//   out[r] += 0.5*(PA[c] + v0[e]*PB0[c] + v1[e]*PB1[c])   for each edge e=(r,c)
//   out = relu(out + bias)
//
// GEMMs use V_WMMA_F32_16X16X4_F32 (exact f32). Weights are pre-packed into the
// WMMA B-operand lane layout so the hot loop does 4 coalesced b128 loads + 8 WMMA
// per K-step. X tiles staged to LDS via GLOBAL_LOAD_ASYNC_TO_LDS_B128.

typedef float v2f __attribute__((ext_vector_type(2)));
typedef float v8f __attribute__((ext_vector_type(8)));

#define F_IN       256
#define F_OUT      128
#define TILE_M     16
#define LDS_STRIDE 260   // 256 + 4 pad: rows stay 16B-aligned, kills LDS bank conflicts
#define WSZ        (F_IN * F_OUT)   // 32768 floats per weight matrix

#if __has_builtin(__builtin_amdgcn_global_load_async_to_lds_b128)
#define HAVE_ASYNC_LDS 1
// Builtin prototype (from hipcc diagnostic): param0 is
//   __attribute__((vector_size(16))) int __device__ *   (addrspace(1))
typedef int v4i_vs __attribute__((vector_size(4 * sizeof(int))));
typedef __attribute__((address_space(1))) v4i_vs* gptr128_t;
typedef __attribute__((address_space(3))) v4i_vs* lptr128_t;
#else
#define HAVE_ASYNC_LDS 0
#endif

// ---------------------------------------------------------------------------
// Pre-pack weights into WMMA B-operand order (and pre-sum W_A0+W_A1).
// Packed group g = k0q*4 + tp holds, for wave lane l (l = slot):
//   float4 at wp[g*128 + l*4] = { W(k,n0), W(k+1,n0), W(k,n1), W(k+1,n1) }
//   with m = l&15, kb = 2*(l>>4), k = k0q*4+kb, n0 = 2*tp*16+m, n1 = n0+16.
// This makes the GEMM B fetch one 512B fully-coalesced global_load_b128 per
// (k-step, tile-pair), always L1/L2-resident (128 KB per matrix).
// ---------------------------------------------------------------------------
__global__ __launch_bounds__(256)
void pack_weights(const float* __restrict__ wA,
                  const float* __restrict__ wB,
                  float* __restrict__ wp)
{
    const int which = blockIdx.y;                       // 0:A-sum 1:B0 2:B1
    const int i     = blockIdx.x * 256 + threadIdx.x;   // [0, 8192)
    const int slot  = i & 31;
    const int tp    = (i >> 5) & 3;
    const int k0q   = i >> 7;

    const int m  = slot & 15;
    const int kb = (slot >> 4) << 1;
    const int k  = (k0q << 2) + kb;
    const int n0 = (tp << 5) + m;                       // 2*tp*16 + m
    const int n1 = n0 + 16;

    const float* s0 = (which == 0) ? wA : (which == 1 ? wB : wB + WSZ);
    float4 v;
    v.x = s0[(size_t)k       * F_OUT + n0];
    v.y = s0[(size_t)(k + 1) * F_OUT + n0];
    v.z = s0[(size_t)k       * F_OUT + n1];
    v.w = s0[(size_t)(k + 1) * F_OUT + n1];
    if (which == 0) {
        const float* s1 = wA + WSZ;
        v.x += s1[(size_t)k       * F_OUT + n0];
        v.y += s1[(size_t)(k + 1) * F_OUT + n0];
        v.z += s1[(size_t)k       * F_OUT + n1];
        v.w += s1[(size_t)(k + 1) * F_OUT + n1];
    }
    *(float4*)(wp + (size_t)which * WSZ + (size_t)i * 4) = v;
}

// ---------------------------------------------------------------------------
// One wave (32 lanes) per block: 16(M) x 128(N) f32 output strip via WMMA.
// ---------------------------------------------------------------------------
__global__ __launch_bounds__(32)
void gemm_tile_wmma(const float* __restrict__ X,
                    const float* __restrict__ Wp,    // packed weights (one matrix)
                    float* __restrict__ Y)
{
    __shared__ float xs[TILE_M * LDS_STRIDE];
    const int lane = threadIdx.x;              // 0..31
    const int row0 = blockIdx.x * TILE_M;

    // Stage 16x256 X tile into LDS (fully coalesced float4 granularity).
    for (int i = lane; i < TILE_M * (F_IN / 4); i += 32) {
        const int rr = i >> 6;                 // 64 float4 per row
        const int cc = (i & 63) << 2;
        const float* gsrc = X + (size_t)(row0 + rr) * F_IN + cc;
        float*       ldst = xs + rr * LDS_STRIDE + cc;
#if HAVE_ASYNC_LDS
        __builtin_amdgcn_global_load_async_to_lds_b128(
            (gptr128_t)(const_cast<float*>(gsrc)),
            (lptr128_t)ldst,
            0, 0);
#else
        *(float4*)ldst = *(const float4*)gsrc;
#endif
    }
#if HAVE_ASYNC_LDS
#if __has_builtin(__builtin_amdgcn_s_wait_asynccnt)
    __builtin_amdgcn_s_wait_asynccnt(0);
#else
    asm volatile("s_wait_asynccnt 0x0" ::: "memory");
#endif
#endif
    __syncthreads();

    // WMMA f32 16x16x4 lane layout:
    //   A: lane l holds row M=l&15, K = {kb, kb+1}, kb = 2*(l>>4)
    //   B: lane l holds col N=l&15, same K pair (pre-packed above)
    //   C/D: vgpr j: lanes 0-15 -> M=j, lanes 16-31 -> M=j+8; N = l&15
    const int m  = lane & 15;
    const int kb = (lane >> 4) << 1;

    v8f acc[8];
    #pragma unroll
    for (int t = 0; t < 8; ++t) acc[t] = (v8f){0.f,0.f,0.f,0.f,0.f,0.f,0.f,0.f};

    const float* wrow = Wp + lane * 4;
    for (int k0q = 0; k0q < F_IN / 4; ++k0q) {
        const float2 a2 = *(const float2*)(xs + m * LDS_STRIDE + (k0q << 2) + kb);
        v2f A; A.x = a2.x; A.y = a2.y;

        #pragma unroll
        for (int tp = 0; tp < 4; ++tp) {
            const float4 b = *(const float4*)(wrow + (((k0q << 2) + tp) << 7));
            v2f B0; B0.x = b.x; B0.y = b.y;
            v2f B1; B1.x = b.z; B1.y = b.w;
            acc[2*tp]     = __builtin_amdgcn_wmma_f32_16x16x4_f32(
                false, A, false, B0, (short)0, acc[2*tp],     false, false);
            acc[2*tp + 1] = __builtin_amdgcn_wmma_f32_16x16x4_f32(
                false, A, false, B1, (short)0, acc[2*tp + 1], false, false);
        }
    }

    // Write D per documented 16x16 f32 layout.
    const int mbase = row0 + ((lane >> 4) << 3);
    #pragma unroll
    for (int t = 0; t < 8; ++t) {
        #pragma unroll
        for (int j = 0; j < 8; ++j) {
            Y[(size_t)(mbase + j) * F_OUT + t * 16 + m] = acc[t][j];
        }
    }
}

__global__ __launch_bounds__(256)
void zero_out_k(float* __restrict__ out, int total)
{
    const int i = blockIdx.x * blockDim.x + threadIdx.x;
    if (i < total) out[i] = 0.f;
}

// One wave per edge; each lane owns 4 consecutive output features (float4).
// Gathers hit L2 (3 tables = 77 MB < 192 MB L2); scatter via hardware f32 atomics.
__global__ __launch_bounds__(256)
void edge_accum(const float* __restrict__ sv,      // [2, E]
                const int*   __restrict__ rows,
                const int*   __restrict__ cols,
                const float* __restrict__ PA,
                const float* __restrict__ PB0,
                const float* __restrict__ PB1,
                float* __restrict__ out, int E)
{
    const int gid  = blockIdx.x * blockDim.x + threadIdx.x;
    const int e    = gid >> 5;
    const int lane = gid & 31;
    if (e >= E) return;

    const int   r  = rows[e];
    const int   c  = cols[e];
    const float v0 = sv[e];
    const float v1 = sv[E + e];

    const int f = lane << 2;
    const float4 pa = *(const float4*)(PA  + (size_t)c * F_OUT + f);
    const float4 b0 = *(const float4*)(PB0 + (size_t)c * F_OUT + f);
    const float4 b1 = *(const float4*)(PB1 + (size_t)c * F_OUT + f);

    float* o = out + (size_t)r * F_OUT + f;
    unsafeAtomicAdd(o + 0, 0.5f * (pa.x + v0 * b0.x + v1 * b1.x));
    unsafeAtomicAdd(o + 1, 0.5f * (pa.y + v0 * b0.y + v1 * b1.y));
    unsafeAtomicAdd(o + 2, 0.5f * (pa.z + v0 * b0.z + v1 * b1.z));
    unsafeAtomicAdd(o + 3, 0.5f * (pa.w + v0 * b0.w + v1 * b1.w));
}

__global__ __launch_bounds__(256)
void finalize_k(float* __restrict__ out, const float* __restrict__ bias, int total)
{
    const int i = blockIdx.x * blockDim.x + threadIdx.x;
    if (i < total) {
        const float v = out[i] + bias[i & (F_OUT - 1)];
        out[i] = v > 0.f ? v : 0.f;
    }
}

extern "C" void kernel_launch(void* const* d_in, const int* in_sizes, int n_in,
                              void* d_out, int out_size, void* d_ws, size_t ws_size,
                              hipStream_t stream)
{
    const float* x    = (const float*)d_in[0];   // [N, 256]
    const float* sv   = (const float*)d_in[1];   // [2, E]
    const float* wA   = (const float*)d_in[2];   // [2, 256, 128]
    const float* wB   = (const float*)d_in[3];   // [2, 256, 128]
    const float* bias = (const float*)d_in[4];   // [128]
    const int*   er   = (const int*)d_in[5];     // [E]
    const int*   ec   = (const int*)d_in[6];     // [E]

    float* out = (float*)d_out;
    float* ws  = (float*)d_ws;

    const int    Nn = in_sizes[0] / F_IN;        // 50000 (multiple of 16)
    const int    E  = in_sizes[5];               // 600000
    const size_t NF = (size_t)Nn * F_OUT;

    float* PA  = ws;
    float* PB0 = ws + NF;
    float* PB1 = ws + 2 * NF;
    float* wp  = ws + 3 * NF;                    // 3 packed matrices, 384 KB

    const int total = Nn * F_OUT;
    zero_out_k<<<(total + 255) / 256, 256, 0, stream>>>(out, total);

    pack_weights<<<dim3(8192 / 256, 3), 256, 0, stream>>>(wA, wB, wp);

    const dim3 ggrid(Nn / TILE_M);               // 3125 tiles
    gemm_tile_wmma<<<ggrid, 32, 0, stream>>>(x, wp,           PA);
    gemm_tile_wmma<<<ggrid, 32, 0, stream>>>(x, wp + WSZ,     PB0);
    gemm_tile_wmma<<<ggrid, 32, 0, stream>>>(x, wp + 2 * WSZ, PB1);

    const long ethreads = (long)E * 32;
    edge_accum<<<(unsigned)((ethreads + 255) / 256), 256, 0, stream>>>(
        sv, er, ec, PA, PB0, PB1, out, E);

    finalize_k<<<(total + 255) / 256, 256, 0, stream>>>(out, bias, total);
}